// Butterfly_44049184588441
// MI455X (gfx1250) — compile-verified
//
#include <hip/hip_runtime.h>

typedef __attribute__((ext_vector_type(2))) float v2f;
typedef __attribute__((ext_vector_type(4))) float v4f;
typedef __attribute__((ext_vector_type(8))) float v8f;

#define BFN     2048
#define TB      16          // batch rows per workgroup tile
#define LSTR    17          // padded LDS stride (floats) per component row
#define WMATF   256         // floats per 16x16 matrix, fragment-ordered

// 2x2 butterfly: a' = t00*a + t01*b ; b' = t10*a + t11*b  (t = {t00,t01,t10,t11})
#define BFLY(a, b, t)                                  \
  do {                                                 \
    float _ta = (a), _tb = (b);                        \
    (a) = fmaf((t).x, _ta, (t).y * _tb);               \
    (b) = fmaf((t).z, _ta, (t).w * _tb);               \
  } while (0)

// ---------------------------------------------------------------------------
// Precompute: compose stages {3..6} and {7..10} into dense 16x16 matrices.
//   matId 0..127   : G1, positions 128*Bblk + r + 8*k   (Bblk=matId>>3, r=matId&7)
//   matId 128..255 : G2, positions p + 128*k            (p=matId-128)
// One thread per (matrix, column). Output is stored in WMMA A-fragment order:
//   W[mat][chunk(0..3)][lane(0..31)][2]  with lane L: rows, lane halves hold K,K+2.
// ---------------------------------------------------------------------------
__global__ __launch_bounds__(256) void bf_precompute(const float* __restrict__ tw,
                                                     float* __restrict__ W) {
  int gid = blockIdx.x * 256 + threadIdx.x;   // 4096 threads
  int matId = gid >> 4;
  int col = gid & 15;
  if (matId >= 256) return;

  float v[16];
#pragma unroll
  for (int i = 0; i < 16; ++i) v[i] = (i == col) ? 1.0f : 0.0f;

  int base, step, idx0;
  if (matId < 128) { base = 128 * (matId >> 3) + (matId & 7); step = 8;   idx0 = 3; }
  else             { base = matId - 128;                      step = 128; idx0 = 7; }

  for (int st = 0; st < 4; ++st) {
    int idx = idx0 + st;
    int sk = 1 << st;            // stride in k-units
    int s = step * sk;           // global stride of this stage
    for (int pb = 0; pb < 8; ++pb) {
      int grp = pb >> st;
      int within = pb & (sk - 1);
      int k0 = grp * 2 * sk + within;
      int k1 = k0 + sk;
      int e0 = base + step * k0;          // global position of j=0 element
      int gi = e0 / (2 * s);
      int sp = e0 % (2 * s);              // guaranteed < s
      int tp = gi * s + sp;               // twiddle pair index
      const float* t = tw + ((idx * 1024) + tp) * 4;
      float a = v[k0], b = v[k1];
      v[k0] = t[0] * a + t[1] * b;
      v[k1] = t[2] * a + t[3] * b;
    }
  }

  // scatter column into fragment order (k = col is the contraction index)
  int chunk = col >> 2;
  int kk = col & 3;
  int kh = kk >> 1;
  int slot = kk & 1;
  float* Wm = W + matId * WMATF;
#pragma unroll
  for (int i = 0; i < 16; ++i)
    Wm[(chunk * 32 + (i + 16 * kh)) * 2 + slot] = v[i];
}

// ---------------------------------------------------------------------------
// One 16x16 x 16x16 tile: D(comps x batch) = M(16x16) * X(comps x batch), in-place
// in LDS. K=16 split into 4 chunks of V_WMMA_F32_16X16X4_F32.
// ---------------------------------------------------------------------------
__device__ __forceinline__ void wmma_tile(float* lds, const float* __restrict__ Wm,
                                          int posBase, int posStep, int lane) {
  const int n = lane & 15;        // batch column (N)
  const int kh = lane >> 4;       // lane-half: K offset +2 / D rows +8
  v2f A[4], B[4];
#pragma unroll
  for (int ch = 0; ch < 4; ++ch) {
    A[ch] = *(const v2f*)(Wm + ch * 64 + lane * 2);         // coalesced 8B/lane
    int k0 = ch * 4 + 2 * kh;
    B[ch].x = lds[(posBase + posStep * (k0 + 0)) * LSTR + n];
    B[ch].y = lds[(posBase + posStep * (k0 + 1)) * LSTR + n];
  }
  v8f c = {0.f, 0.f, 0.f, 0.f, 0.f, 0.f, 0.f, 0.f};
#pragma unroll
  for (int ch = 0; ch < 4; ++ch)
    c = __builtin_amdgcn_wmma_f32_16x16x4_f32(false, A[ch], false, B[ch],
                                              (short)0, c, false, false);
#pragma unroll
  for (int vv = 0; vv < 8; ++vv) {
    int mrow = vv + 8 * kh;       // VGPR v: rows v (lanes 0-15) / v+8 (lanes 16-31)
    lds[(posBase + posStep * mrow) * LSTR + n] = c[vv];
  }
}

// ---------------------------------------------------------------------------
// Main: per workgroup, 16 batch rows x 2048 cols.
//   G0: global->regs, butterfly stages 0..2 (contiguous 8-blocks), regs->LDS (transposed)
//   G1: 128 WMMA tiles (stride-8 subvectors inside 128-blocks)
//   G2: 128 WMMA tiles (stride-128 subvectors across the row)
//   out: LDS + bias -> global
// ---------------------------------------------------------------------------
__global__ __launch_bounds__(256) void bf_main(const float* __restrict__ x,
                                               const float* __restrict__ tw,
                                               const float* __restrict__ bias,
                                               const float* __restrict__ W,
                                               float* __restrict__ out) {
  __shared__ float lds[BFN * LSTR];       // 2048*17*4 = 136 KB

  const int tid = threadIdx.x;
  const int lane = tid & 31;
  const int wave = tid >> 5;
  const long rowBase = (long)blockIdx.x * TB;

  // ---- G0: stages 0..2 in registers while loading -------------------------
  {
    const int m = tid;                    // 8-element block 0..255
    v4f tq[12];                           // tq[idx*4+q] = {t00,t01,t10,t11}
#pragma unroll
    for (int idx = 0; idx < 3; ++idx)
#pragma unroll
      for (int q = 0; q < 4; ++q)
        tq[idx * 4 + q] = *(const v4f*)(tw + idx * 4096 + m * 16 + q * 4);

    for (int rb = 0; rb < TB; ++rb) {
      const float* xr = x + (rowBase + rb) * BFN + m * 8;
      v4f a0 = *(const v4f*)(xr);
      v4f a1 = *(const v4f*)(xr + 4);
      float v0 = a0.x, v1 = a0.y, v2 = a0.z, v3 = a0.w;
      float v4 = a1.x, v5 = a1.y, v6 = a1.z, v7 = a1.w;
      // stage 0 (s=1): pairs (0,1)(2,3)(4,5)(6,7), twiddle q = pair
      BFLY(v0, v1, tq[0]); BFLY(v2, v3, tq[1]);
      BFLY(v4, v5, tq[2]); BFLY(v6, v7, tq[3]);
      // stage 1 (s=2): pairs (0,2)(1,3)(4,6)(5,7)
      BFLY(v0, v2, tq[4]); BFLY(v1, v3, tq[5]);
      BFLY(v4, v6, tq[6]); BFLY(v5, v7, tq[7]);
      // stage 2 (s=4): pairs (0,4)(1,5)(2,6)(3,7)
      BFLY(v0, v4, tq[8]);  BFLY(v1, v5, tq[9]);
      BFLY(v2, v6, tq[10]); BFLY(v3, v7, tq[11]);
      // transpose-store: component-major
      float vs[8] = {v0, v1, v2, v3, v4, v5, v6, v7};
#pragma unroll
      for (int j = 0; j < 8; ++j)
        lds[(m * 8 + j) * LSTR + rb] = vs[j];
    }
  }
  __syncthreads();

  // ---- G1: stages 3..6 via WMMA ------------------------------------------
  for (int i = 0; i < 16; ++i) {
    int t = wave + 8 * i;                          // 0..127
    int posBase = ((t >> 3) << 7) + (t & 7);       // 128*Bblk + r
    wmma_tile(lds, W + t * WMATF, posBase, 8, lane);
  }
  __syncthreads();

  // ---- G2: stages 7..10 via WMMA -----------------------------------------
  for (int i = 0; i < 16; ++i) {
    int t = wave + 8 * i;                          // p = 0..127
    wmma_tile(lds, W + (128 + t) * WMATF, t, 128, lane);
  }
  __syncthreads();

  // ---- bias + store -------------------------------------------------------
  {
    const int m = tid;
    v4f b0 = *(const v4f*)(bias + m * 8);
    v4f b1 = *(const v4f*)(bias + m * 8 + 4);
    for (int rb = 0; rb < TB; ++rb) {
      v4f o0, o1;
      o0.x = lds[(m * 8 + 0) * LSTR + rb] + b0.x;
      o0.y = lds[(m * 8 + 1) * LSTR + rb] + b0.y;
      o0.z = lds[(m * 8 + 2) * LSTR + rb] + b0.z;
      o0.w = lds[(m * 8 + 3) * LSTR + rb] + b0.w;
      o1.x = lds[(m * 8 + 4) * LSTR + rb] + b1.x;
      o1.y = lds[(m * 8 + 5) * LSTR + rb] + b1.y;
      o1.z = lds[(m * 8 + 6) * LSTR + rb] + b1.z;
      o1.w = lds[(m * 8 + 7) * LSTR + rb] + b1.w;
      float* orow = out + (rowBase + rb) * BFN + m * 8;
      *(v4f*)(orow) = o0;
      *(v4f*)(orow + 4) = o1;
    }
  }
}

extern "C" void kernel_launch(void* const* d_in, const int* in_sizes, int n_in,
                              void* d_out, int out_size, void* d_ws, size_t ws_size,
                              hipStream_t stream) {
  (void)in_sizes; (void)n_in; (void)out_size; (void)ws_size;
  const float* x    = (const float*)d_in[0];   // (16384, 2048)
  const float* tw   = (const float*)d_in[1];   // (1,1,11,1024,2,2)
  const float* bias = (const float*)d_in[2];   // (2048,)
  float* outp = (float*)d_out;
  float* W = (float*)d_ws;                     // 256 matrices * 256 floats = 256 KB

  bf_precompute<<<16, 256, 0, stream>>>(tw, W);
  bf_main<<<16384 / TB, 256, 0, stream>>>(x, tw, bias, W, outp);
}